// GAT_75952201662935
// MI455X (gfx1250) — compile-verified
//
#include <hip/hip_runtime.h>

// ---------------------------------------------------------------- constants
#define N_NODES 50000
#define N_EDGES 800000
#define EP      (N_EDGES + N_NODES)   /* edges + self loops = 850000 */
#define NGROUP  64
#define HID     64
#define HEADS   4
#define CH      16
#define OUTF    5
#define NEG_SLOPE 0.2f

typedef __attribute__((ext_vector_type(2))) float v2f;
typedef __attribute__((ext_vector_type(8))) float v8f;

// Order-preserving float<->uint encoding for atomic max on signed floats.
__device__ __forceinline__ unsigned ord_f32(float f) {
  unsigned u = __float_as_uint(f);
  return (u & 0x80000000u) ? ~u : (u | 0x80000000u);
}
__device__ __forceinline__ float unord_f32(unsigned u) {
  return (u & 0x80000000u) ? __uint_as_float(u & 0x7FFFFFFFu)
                           : __uint_as_float(~u);
}

// ---------------------------------------------------------------- utilities
__global__ void zero_f32_kernel(float* __restrict__ p, int n) {
  int t = blockIdx.x * blockDim.x + threadIdx.x;
  if (t < n) p[t] = 0.0f;
}

// ------------------------------------------------------- layer 0: F_IN == 1
// feat[n, j] = x[n] * W0[j]
__global__ void feat0_kernel(const float* __restrict__ x,
                             const float* __restrict__ W0,
                             float* __restrict__ feat) {
  int t = blockIdx.x * blockDim.x + threadIdx.x;
  if (t >= N_NODES * HID) return;
  int n = t >> 6, j = t & 63;
  feat[t] = x[n] * W0[j];
}

// ------------------------------------------- fp32 WMMA GEMM: C = A[rows,64] @ B[64,64]
// One wave computes a 16x64 tile: 4 accumulators of 16x16, K stepped by 4
// with V_WMMA_F32_16X16X4_F32.
// B is staged once per workgroup into LDS, pre-packed in the WMMA B-fragment
// layout: float2 element [t*64+col] = (B[2t][col], B[2t+1][col]). A lane's
// B fragment for (k, half, nt) is then a single aligned ds_load_b64 directly
// into the WMMA SRC1 register pair (no repacking movs); the 4 nt fragments
// are 128B apart so pairs fuse into ds_load_2addr_b64.
// f32 A (16x4) fragment layout: lanes 0-15 -> M=0..15 K={0,1}; lanes 16-31 -> K={2,3}.
// f32 B (4x16) fragment layout: lanes 0-15 -> N=0..15 K={0,1}; lanes 16-31 -> K={2,3}.
// f32 D (16x16): VGPR r, lanes 0-15 -> M=r, lanes 16-31 -> M=8+r, N=lane&15.
__global__ __launch_bounds__(256) void gemm_n64_wmma(
    const float* __restrict__ A, const float* __restrict__ B,
    float* __restrict__ C, int nrows) {
  __shared__ float2 Bs2[(HID * HID) / 2];    // 16 KB of 320 KB/WGP

  // cooperative fill in fragment-pair layout: 2048 float2, 256 threads -> 8 each
  for (int p = threadIdx.x; p < (HID * HID) / 2; p += 256) {
    int t = p >> 6, col = p & 63;
    float2 v;
    v.x = B[(2 * t)     * HID + col];
    v.y = B[(2 * t + 1) * HID + col];
    Bs2[p] = v;
  }
  __syncthreads();                           // before any early return!

  const int lane = threadIdx.x & 31;
  const int wave = blockIdx.x * (blockDim.x >> 5) + (threadIdx.x >> 5);
  const int row0 = wave * 16;
  if (row0 >= nrows) return;                 // wave-uniform; EXEC all-ones inside
  const int half = lane >> 4;                // 0: K lo pair, 1: K hi pair
  const int l    = lane & 15;
  const int arow = row0 + l;

  v8f acc[4] = {v8f{0,0,0,0,0,0,0,0}, v8f{0,0,0,0,0,0,0,0},
                v8f{0,0,0,0,0,0,0,0}, v8f{0,0,0,0,0,0,0,0}};

#pragma unroll
  for (int k = 0; k < HID; k += 4) {
    const int kk = k + half * 2;             // even -> float2-aligned
    const float2 a2 = *(const float2*)(A + arow * HID + kk);
    v2f a;  a.x = a2.x;  a.y = a2.y;
    const int tpair = (k >> 1) + half;       // K-pair index into packed LDS
#pragma unroll
    for (int nt = 0; nt < 4; ++nt) {
      const float2 b2 = Bs2[tpair * HID + nt * 16 + l];
      v2f b;  b.x = b2.x;  b.y = b2.y;
      acc[nt] = __builtin_amdgcn_wmma_f32_16x16x4_f32(
          false, a, false, b, (short)0, acc[nt], false, false);
    }
  }

  const int mbase = row0 + half * 8;
#pragma unroll
  for (int nt = 0; nt < 4; ++nt)
#pragma unroll
    for (int r = 0; r < 8; ++r)
      C[(mbase + r) * HID + nt * 16 + l] = acc[nt][r];
}

// -------------------------------- per-node attention coefficients (per head)
__global__ void alpha_kernel(const float* __restrict__ feat,
                             const float* __restrict__ a_src,
                             const float* __restrict__ a_dst,
                             float* __restrict__ as, float* __restrict__ ad) {
  int i = blockIdx.x * blockDim.x + threadIdx.x;
  if (i >= N_NODES) return;
  const float* f = feat + i * HID;
#pragma unroll
  for (int h = 0; h < HEADS; ++h) {
    float s = 0.f, d = 0.f;
#pragma unroll
    for (int c = 0; c < CH; ++c) {
      float v = f[h * CH + c];
      s += v * a_src[h * CH + c];
      d += v * a_dst[h * CH + c];
    }
    as[i * HEADS + h] = s;
    ad[i * HEADS + h] = d;
  }
}

// ------------------------------ edge pass 1: logits + segment max (per head)
__global__ __launch_bounds__(256) void edge_pass1(
    const int* __restrict__ ei, const float* __restrict__ as,
    const float* __restrict__ ad, float* __restrict__ ebuf,
    unsigned* __restrict__ m) {
  int t = blockIdx.x * blockDim.x + threadIdx.x;
  if (t >= EP * HEADS) return;
  int eid = t >> 2, h = t & 3;
  int s, d;
  if (eid < N_EDGES) { s = ei[eid]; d = ei[N_EDGES + eid]; }
  else               { s = d = eid - N_EDGES; }            // self loop
  float v = as[s * HEADS + h] + ad[d * HEADS + h];
  v = v > 0.f ? v : NEG_SLOPE * v;                          // leaky relu
  ebuf[eid * HEADS + h] = v;
  atomicMax(&m[d * HEADS + h], ord_f32(v));
}

// ---- edge pass 2: w=exp(e-max); z[dst]+=w; hout[dst]+= w*feat[src] (64 ch)
__global__ __launch_bounds__(256) void edge_pass2(
    const int* __restrict__ ei, const float* __restrict__ ebuf,
    const unsigned* __restrict__ m, const float* __restrict__ feat,
    float* __restrict__ z, float* __restrict__ hout) {
  int t = blockIdx.x * blockDim.x + threadIdx.x;
  if (t >= EP * HID) return;                  // 54.4M threads
  int eid = t >> 6;
  int lane = t & 63, h = lane >> 4, c = lane & 15;
  int s, d;
  if (eid < N_EDGES) { s = ei[eid]; d = ei[N_EDGES + eid]; }
  else               { s = d = eid - N_EDGES; }
  float w = __expf(ebuf[eid * HEADS + h] - unord_f32(m[d * HEADS + h]));
  if (c == 0) atomicAdd(&z[d * HEADS + h], w);
  atomicAdd(&hout[d * HID + h * CH + c], w * feat[s * HID + h * CH + c]);
}

// ------------------------- node finalize: divide by softmax denom, add bias
__global__ void node_finalize(float* __restrict__ h, const float* __restrict__ z,
                              const float* __restrict__ bias) {
  int t = blockIdx.x * blockDim.x + threadIdx.x;
  if (t >= N_NODES * HID) return;
  int i = t >> 6, j = t & 63;
  h[t] = h[t] / z[i * HEADS + (j >> 4)] + bias[j];
}

// ------------------------------------------------------------- mean pooling
__global__ void pool_acc(const float* __restrict__ h, const int* __restrict__ batch,
                         float* __restrict__ psum, float* __restrict__ pcnt) {
  int t = blockIdx.x * blockDim.x + threadIdx.x;
  if (t >= N_NODES * HID) return;
  int i = t >> 6, j = t & 63;
  int g = batch[i];
  atomicAdd(&psum[g * HID + j], h[t]);
  if (j == 0) atomicAdd(&pcnt[g], 1.0f);
}

__global__ void final_mm(const float* __restrict__ psum, const float* __restrict__ pcnt,
                         const float* __restrict__ Wf, const float* __restrict__ bf,
                         float* __restrict__ out) {
  int t = blockIdx.x * blockDim.x + threadIdx.x;
  if (t >= NGROUP * OUTF) return;
  int g = t / OUTF, o = t % OUTF;
  float inv = 1.0f / fmaxf(pcnt[g], 1.0f);
  float acc = bf[o];
#pragma unroll
  for (int j = 0; j < HID; ++j) acc += psum[g * HID + j] * inv * Wf[j * OUTF + o];
  out[t] = acc;
}

// ---------------------------------------------------------------- launcher
static inline int cdiv(long long a, long long b) { return (int)((a + b - 1) / b); }

extern "C" void kernel_launch(void* const* d_in, const int* in_sizes, int n_in,
                              void* d_out, int out_size, void* d_ws, size_t ws_size,
                              hipStream_t stream) {
  (void)in_sizes; (void)n_in; (void)out_size; (void)ws_size;
  const float* x    = (const float*)d_in[0];
  const int*   ei   = (const int*)d_in[1];   // [2,E] int32, src row then dst row
  const int*   batch= (const int*)d_in[2];
  const float* W0   = (const float*)d_in[3];
  const float* as0  = (const float*)d_in[4];
  const float* ad0  = (const float*)d_in[5];
  const float* b0   = (const float*)d_in[6];
  const float* Wh   = (const float*)d_in[7];   // [3,64,64]
  const float* ash  = (const float*)d_in[8];   // [3,4,16]
  const float* adh  = (const float*)d_in[9];
  const float* bh   = (const float*)d_in[10];  // [3,64]
  const float* Wf   = (const float*)d_in[11];  // [64,5]
  const float* bf   = (const float*)d_in[12];
  float* out = (float*)d_out;

  // workspace layout (all f32-sized slots)
  const long long NF = (long long)N_NODES * HID;   // 3.2M
  const long long NH = (long long)N_NODES * HEADS; // 200K
  float* ws    = (float*)d_ws;
  float* feat  = ws;                    // NF
  float* hbuf0 = feat + NF;             // NF
  float* hbuf1 = hbuf0 + NF;            // NF
  float* as    = hbuf1 + NF;            // NH
  float* ad    = as + NH;               // NH
  unsigned* mU = (unsigned*)(ad + NH);  // NH (uint)
  float* z     = (float*)mU + NH;       // NH
  float* ebuf  = z + NH;                // EP*HEADS = 3.4M
  float* psum  = ebuf + (long long)EP * HEADS;   // 64*64
  float* pcnt  = psum + NGROUP * HID;            // 64

  const int B = 256;

  for (int layer = 0; layer < 4; ++layer) {
    const float* hin  = (layer == 0) ? nullptr : ((layer & 1) ? hbuf0 : hbuf1);
    float*       hout = (layer & 1) ? hbuf1 : hbuf0;
    const float* a_s  = (layer == 0) ? as0 : ash + (layer - 1) * HEADS * CH;
    const float* a_d  = (layer == 0) ? ad0 : adh + (layer - 1) * HEADS * CH;
    const float* bias = (layer == 0) ? b0  : bh  + (layer - 1) * HID;

    // feat = h_in @ W
    if (layer == 0) {
      feat0_kernel<<<cdiv(NF, B), B, 0, stream>>>(x, W0, feat);
    } else {
      const float* W = Wh + (long long)(layer - 1) * HID * HID;
      int waves = N_NODES / 16;                      // 3125 (exact)
      gemm_n64_wmma<<<cdiv(waves, B / 32), B, 0, stream>>>(hin, W, feat, N_NODES);
    }

    // per-node attention coefficients
    alpha_kernel<<<cdiv(N_NODES, B), B, 0, stream>>>(feat, a_s, a_d, as, ad);

    // reset accumulators: m (ord-encoded, 0 == -inf), z, hout
    zero_f32_kernel<<<cdiv(NH, B), B, 0, stream>>>((float*)mU, (int)NH);
    zero_f32_kernel<<<cdiv(NH, B), B, 0, stream>>>(z, (int)NH);
    zero_f32_kernel<<<cdiv(NF, B), B, 0, stream>>>(hout, (int)NF);

    // edge phase
    edge_pass1<<<cdiv((long long)EP * HEADS, B), B, 0, stream>>>(ei, as, ad, ebuf, mU);
    edge_pass2<<<cdiv((long long)EP * HID, B), B, 0, stream>>>(ei, ebuf, mU, feat, z, hout);

    // normalize + bias
    node_finalize<<<cdiv(NF, B), B, 0, stream>>>(hout, z, bias);
  }

  // global mean pool over groups, then [64,5] linear
  zero_f32_kernel<<<cdiv(NGROUP * HID + NGROUP, B), B, 0, stream>>>(psum, NGROUP * HID + NGROUP);
  pool_acc<<<cdiv(NF, B), B, 0, stream>>>(hbuf1, batch, psum, pcnt);
  final_mm<<<1, 512, 0, stream>>>(psum, pcnt, Wf, bf, out);
}